// ConstantODEblock1_57535381897865
// MI455X (gfx1250) — compile-verified
//
#include <hip/hip_runtime.h>
#include <math.h>

#define FEAT_D 32

typedef __attribute__((ext_vector_type(2))) float v2f;
typedef __attribute__((ext_vector_type(8))) float v8f;

// ---------------------------------------------------------------------------
// Zero the scatter accumulator (N*D floats) with 128-bit stores.
// ---------------------------------------------------------------------------
__global__ __launch_bounds__(256)
void zero_kernel(float4* __restrict__ ax4, int n4) {
    int i = blockIdx.x * blockDim.x + threadIdx.x;
    if (i < n4) ax4[i] = make_float4(0.f, 0.f, 0.f, 0.f);
}

// ---------------------------------------------------------------------------
// Scatter: one wave32 per edge, lane = feature dim.
//   - src/dst/w are wave-uniform -> force SGPR via readfirstlane (s_load path)
//   - x[src] row read is a single coalesced 128B wave load
//   - ax[dst] row update is a coalesced 128B row of global_atomic_add_f32
//   - speculative global_prefetch_b8 on the streaming edge arrays
// ---------------------------------------------------------------------------
__global__ __launch_bounds__(256)
void scatter_kernel(const float* __restrict__ x,
                    const int*   __restrict__ src,
                    const int*   __restrict__ dst,
                    const float* __restrict__ w,
                    float*       __restrict__ ax,
                    int E) {
    int wave = blockIdx.x * (blockDim.x >> 5) + (threadIdx.x >> 5);
    int e = __builtin_amdgcn_readfirstlane(wave);   // wave-uniform edge id
    if (e >= E) return;                             // uniform exit, EXEC stays full

    // Prefetch the edge stream ~32KB ahead (speculative: OOB silently dropped).
    __builtin_prefetch(&src[e + 8192], 0, 1);
    __builtin_prefetch(&dst[e + 8192], 0, 1);
    __builtin_prefetch(&w[e + 8192], 0, 1);

    int   s  = src[e];
    int   t  = dst[e];
    float we = w[e];

    int lane = threadIdx.x & 31;
    float val = we * x[s * FEAT_D + lane];
    unsafeAtomicAdd(&ax[t * FEAT_D + lane], val);   // native global_atomic_add_f32
}

// ---------------------------------------------------------------------------
// Update: out = (1-alpha)*x + beta*x0 + alpha*ax, per 16x16 f32 tile per wave.
// The alpha*ax term is accumulated on the matrix pipe with 4 chained
// V_WMMA_F32_16X16X4_F32 ops: A_m = alpha * (rows 4m..4m+3 selector),
// B_m = rows 4m..4m+3 of the ax tile, C accumulates.
//
// Layouts (wave32):
//   C/D  : lane L -> col N = L%16 ; VGPR v -> row M = v + 8*(L>=16)
//   A 16x4: lane L -> row M = L%16 ; VGPR v -> K = v + 2*(L>=16)
//   B 4x16: lane L -> col N = L%16 ; VGPR v -> K = v + 2*(L>=16)
// ---------------------------------------------------------------------------
__global__ __launch_bounds__(256)
void update_wmma_kernel(const float* __restrict__ x,
                        const float* __restrict__ ax,
                        const float* __restrict__ x0,
                        const float* __restrict__ alpha_p,
                        const float* __restrict__ beta_p,
                        float*       __restrict__ out,
                        int ntiles) {
    int tile = __builtin_amdgcn_readfirstlane(
        blockIdx.x * (blockDim.x >> 5) + (threadIdx.x >> 5));
    if (tile >= ntiles) return;                     // uniform exit

    const int nb = (tile >> 1) * 16;                // node (row) base
    const int cb = (tile & 1) * 16;                 // feature (col) base

    const int lane = threadIdx.x & 31;
    const int hi   = lane >> 4;                     // lane half
    const int li   = lane & 15;                     // N (B/C/D) or M (A)

    float araw  = *alpha_p;
    float alpha = 1.0f / (1.0f + expf(-araw));      // sigmoid
    float beta  = *beta_p;
    float ca    = 1.0f - alpha;

    // C init: (1-alpha)*x + beta*x0 laid out in C/D register order.
    v8f c;
#pragma unroll
    for (int v = 0; v < 8; ++v) {
        int row = nb + v + 8 * hi;
        int idx = row * FEAT_D + cb + li;
        c[v] = ca * x[idx] + beta * x0[idx];
    }

    // Accumulate alpha * ax_tile via 4 WMMAs (identity-selector A).
#pragma unroll
    for (int m = 0; m < 4; ++m) {
        v2f a, b;
#pragma unroll
        for (int v = 0; v < 2; ++v) {
            int k = v + 2 * hi;                     // K index this VGPR/lane holds
            a[v] = (li == 4 * m + k) ? alpha : 0.0f;
            int rowb = nb + 4 * m + k;              // B_m row = tile row 4m+k
            b[v] = ax[rowb * FEAT_D + cb + li];
        }
        // (neg_a, A, neg_b, B, c_mod, C, reuse_a, reuse_b)
        c = __builtin_amdgcn_wmma_f32_16x16x4_f32(
                false, a, false, b, (short)0, c, false, false);
    }

#pragma unroll
    for (int v = 0; v < 8; ++v) {
        int row = nb + v + 8 * hi;
        out[row * FEAT_D + cb + li] = c[v];
    }
}

// ---------------------------------------------------------------------------
// Orchestration: 4 Euler steps. ws holds only the ax accumulator (12.8 MB);
// the evolving x state lives in d_out (elementwise update is in-place safe).
// ---------------------------------------------------------------------------
extern "C" void kernel_launch(void* const* d_in, const int* in_sizes, int n_in,
                              void* d_out, int out_size, void* d_ws, size_t ws_size,
                              hipStream_t stream) {
    const float* x_in    = (const float*)d_in[0];
    const float* w       = (const float*)d_in[1];
    const float* x0      = (const float*)d_in[2];
    const float* alpha_p = (const float*)d_in[3];
    const float* beta_p  = (const float*)d_in[4];
    const int*   eidx    = (const int*)  d_in[5];

    const int N = in_sizes[0] / FEAT_D;
    const int E = in_sizes[1];
    const int* src = eidx;          // edge_index[0]
    const int* dst = eidx + E;      // edge_index[1]

    float* ax  = (float*)d_ws;
    float* out = (float*)d_out;

    const int ND = N * FEAT_D;
    dim3 blk(256);
    const int zeroBlocks = (ND / 4 + 255) / 256;
    const int scatBlocks = (E + 7) / 8;               // 8 waves (edges) per block
    const int ntiles     = (N / 16) * (FEAT_D / 16);  // 16x16 tiles
    const int updBlocks  = (ntiles + 7) / 8;

    const float* xcur = x_in;
    for (int step = 0; step < 4; ++step) {
        zero_kernel<<<zeroBlocks, blk, 0, stream>>>((float4*)ax, ND / 4);
        scatter_kernel<<<scatBlocks, blk, 0, stream>>>(xcur, src, dst, w, ax, E);
        update_wmma_kernel<<<updBlocks, blk, 0, stream>>>(
            xcur, ax, x0, alpha_p, beta_p, out, ntiles);
        xcur = out;
    }
}